// TwoLayerMicroArchitectureBuild_16784732192997
// MI455X (gfx1250) — compile-verified
//
#include <hip/hip_runtime.h>

// ---------------------------------------------------------------------------
// GCN 2-layer pipeline for MI455X (gfx1250, wave32).
// Memory-bound workload (~180MB working set, L2-resident on 192MB L2).
// GEMMs use V_WMMA_F32_16X16X4_F32 (exact fp32 — GEMMs are <5% of runtime,
// so no precision downcast is warranted; edge scatter dominates).
// ---------------------------------------------------------------------------

typedef __attribute__((ext_vector_type(2))) float v2f;
typedef __attribute__((ext_vector_type(8))) float v8f;

#define N_NODES   100000
#define N_EDGES   1600000
#define F_IN      128
#define H_DIM     64
#define C_OUT     40
#define G_GRAPHS  2048

// ---------------------------------------------------------------- zero fill
__global__ void k_zero(float* __restrict__ p, long n) {
  long i = (long)blockIdx.x * blockDim.x + threadIdx.x;
  if (i < n) p[i] = 0.0f;
}

// ------------------------------------------------------- degree (dst counts)
__global__ void k_deg(const long long* __restrict__ dst, float* __restrict__ deg) {
  int e = blockIdx.x * blockDim.x + threadIdx.x;
  if (e < N_EDGES) atomicAdd(&deg[(int)dst[e]], 1.0f);
}

__global__ void k_rsqrt(const float* __restrict__ deg, float* __restrict__ disq) {
  int i = blockIdx.x * blockDim.x + threadIdx.x;
  if (i < N_NODES) disq[i] = rsqrtf(deg[i] + 1.0f);   // +1 self loop
}

// -------------------------------------------------- WMMA GEMM: [n,K]@[K,64]
// One wave computes a 16x16 output tile via V_WMMA_F32_16X16X4_F32.
// Block = 128 threads = 4 waves -> 16 rows x 64 cols per block.
// A frag (16x4 f32, 2 VGPR): lane l -> m=l&15, kb=(l>>4)*2, holds A[m][kb..kb+1]
// B frag (4x16 f32, 2 VGPR): lane l -> n=l&15, kb=(l>>4)*2, holds B[kb..kb+1][n]
// C/D (8 VGPR): VGPR v, lane l -> D[v + 8*(l>>4)][l&15]
__global__ __launch_bounds__(128)
void k_gemm64(const float* __restrict__ in, const float* __restrict__ W,
              const float* __restrict__ bias, float* __restrict__ out,
              int nrows, int K) {
  const int lane    = threadIdx.x & 31;
  const int wave    = threadIdx.x >> 5;        // 0..3 => column tile
  const int rowBase = blockIdx.x * 16;
  const int colBase = wave * 16;
  const int m  = lane & 15;                    // also the n index for B/C
  const int kb = (lane >> 4) * 2;

  int r0 = rowBase + m;
  int rA = (r0 < nrows) ? r0 : (nrows - 1);    // clamp for tail safety
  const float* arow = in + (size_t)rA * K + kb;

  v8f acc = {};
  for (int k = 0; k < K; k += 4) {
    v2f a = *(const v2f*)(arow + k);           // A[m][k+kb], A[m][k+kb+1]
    v2f b;
    b.x = W[(size_t)(k + kb) * H_DIM + colBase + m];
    b.y = W[(size_t)(k + kb + 1) * H_DIM + colBase + m];
    acc = __builtin_amdgcn_wmma_f32_16x16x4_f32(
        /*neg_a=*/false, a, /*neg_b=*/false, b,
        /*c_mod=*/(short)0, acc, /*reuse_a=*/false, /*reuse_b=*/false);
  }

  const int rowOff = (lane >> 4) * 8;
  const float bv = bias ? bias[colBase + m] : 0.0f;
  #pragma unroll
  for (int v = 0; v < 8; ++v) {
    int r = rowBase + rowOff + v;
    if (r < nrows) out[(size_t)r * H_DIM + colBase + m] = acc[v] + bv;
  }
}

// ---------------------------------------------- edge gather + scaled scatter
// 16 lanes per edge, 4 floats (one float4) per lane.
__global__ __launch_bounds__(256)
void k_edge(const long long* __restrict__ src, const long long* __restrict__ dst,
            const float* __restrict__ hw, const float* __restrict__ disq,
            float* __restrict__ agg) {
  long t = (long)blockIdx.x * blockDim.x + threadIdx.x;
  long e = t >> 4;
  if (e >= N_EDGES) return;
  int f = (int)(t & 15) * 4;
  int s = (int)src[e];
  int d = (int)dst[e];
  float coef = disq[s] * disq[d];
  float4 mv = *(const float4*)(hw + (size_t)s * H_DIM + f);
  float* ap = agg + (size_t)d * H_DIM + f;
  atomicAdd(ap + 0, mv.x * coef);
  atomicAdd(ap + 1, mv.y * coef);
  atomicAdd(ap + 2, mv.z * coef);
  atomicAdd(ap + 3, mv.w * coef);
}

// ------------------- fused: self-loop term + conv bias + LayerNorm + ReLU
// One wave (32 lanes) per node; each lane owns features lane and lane+32.
// In-place safe: agg==outh allowed (each thread rewrites only its own slots).
__global__ __launch_bounds__(256)
void k_selfln(const float* __restrict__ agg, const float* __restrict__ hw,
              const float* __restrict__ disq, const float* __restrict__ cbias,
              const float* __restrict__ gamma, const float* __restrict__ beta,
              float* __restrict__ outh) {
  int lane = threadIdx.x & 31;
  int wave = threadIdx.x >> 5;
  int node = blockIdx.x * 8 + wave;
  if (node >= N_NODES) return;
  float di = disq[node];
  float selfc = di * di;                        // deg_isqrt^2 = 1/(deg+1)
  size_t base = (size_t)node * H_DIM;

  float v0 = agg[base + lane]      + hw[base + lane]      * selfc + cbias[lane];
  float v1 = agg[base + lane + 32] + hw[base + lane + 32] * selfc + cbias[lane + 32];

  float s = v0 + v1;
  #pragma unroll
  for (int off = 16; off > 0; off >>= 1) s += __shfl_xor(s, off, 32);
  float mu = s * (1.0f / 64.0f);

  float d0 = v0 - mu, d1 = v1 - mu;
  float q = d0 * d0 + d1 * d1;
  #pragma unroll
  for (int off = 16; off > 0; off >>= 1) q += __shfl_xor(q, off, 32);
  float rs = rsqrtf(q * (1.0f / 64.0f) + 1e-5f);

  float o0 = d0 * rs * gamma[lane]      + beta[lane];
  float o1 = d1 * rs * gamma[lane + 32] + beta[lane + 32];
  outh[base + lane]      = fmaxf(o0, 0.0f);
  outh[base + lane + 32] = fmaxf(o1, 0.0f);
}

// ------------------------------------------ skip sum + segment-sum pooling
__global__ __launch_bounds__(256)
void k_readout(const float* __restrict__ h1, const float* __restrict__ h2,
               const long long* __restrict__ batch, float* __restrict__ pooled) {
  long t = (long)blockIdx.x * blockDim.x + threadIdx.x;
  int node = (int)(t >> 6);
  int f = (int)(t & 63);
  if (node >= N_NODES) return;
  int g = (int)batch[node];
  size_t base = (size_t)node * H_DIM + f;
  atomicAdd(&pooled[(size_t)g * H_DIM + f], h1[base] + h2[base]);
}

// ------------------------------------------------------- head: [G,64]@[64,40]
__global__ __launch_bounds__(256)
void k_post(const float* __restrict__ pooled, const float* __restrict__ W,
            const float* __restrict__ b, float* __restrict__ out) {
  int t = blockIdx.x * blockDim.x + threadIdx.x;
  if (t >= G_GRAPHS * C_OUT) return;
  int g = t / C_OUT, c = t % C_OUT;
  float s = b[c];
  const float* pg = pooled + (size_t)g * H_DIM;
  #pragma unroll 8
  for (int k = 0; k < H_DIM; ++k) s += pg[k] * W[k * C_OUT + c];
  out[t] = s;
}

// ---------------------------------------------------------------------------
extern "C" void kernel_launch(void* const* d_in, const int* in_sizes, int n_in,
                              void* d_out, int out_size, void* d_ws, size_t ws_size,
                              hipStream_t stream) {
  (void)in_sizes; (void)n_in; (void)out_size; (void)ws_size;

  const float*     x      = (const float*)d_in[0];
  const long long* eidx   = (const long long*)d_in[1];   // [2,E] int64
  const long long* batch  = (const long long*)d_in[2];   // [N]   int64
  const float*     pre_w  = (const float*)d_in[3];
  const float*     pre_b  = (const float*)d_in[4];
  const float*     c1_w   = (const float*)d_in[5];
  const float*     c1_b   = (const float*)d_in[6];
  const float*     n1_g   = (const float*)d_in[7];
  const float*     n1_b   = (const float*)d_in[8];
  const float*     c2_w   = (const float*)d_in[9];
  const float*     c2_b   = (const float*)d_in[10];
  const float*     n2_g   = (const float*)d_in[11];
  const float*     n2_b   = (const float*)d_in[12];
  const float*     post_w = (const float*)d_in[13];
  const float*     post_b = (const float*)d_in[14];
  float* out = (float*)d_out;

  const long long* src = eidx;
  const long long* dst = eidx + N_EDGES;

  // Workspace layout (floats). Zeroed region first (deg | bufA | bufB | pooled).
  float* ws     = (float*)d_ws;
  float* deg    = ws;                                   // N
  float* bufA   = deg  + N_NODES;                       // N*64  (agg1 -> h1)
  float* bufB   = bufA + (size_t)N_NODES * H_DIM;       // N*64  (agg2 -> h2)
  float* pooled = bufB + (size_t)N_NODES * H_DIM;       // G*64
  float* disq   = pooled + (size_t)G_GRAPHS * H_DIM;    // N
  float* h      = disq + N_NODES;                       // N*64
  float* hw     = h + (size_t)N_NODES * H_DIM;          // N*64

  const long zeroLen = (long)N_NODES + 2L * N_NODES * H_DIM + (long)G_GRAPHS * H_DIM;

  // 0) zero accumulators (deterministic each call)
  k_zero<<<(unsigned)((zeroLen + 255) / 256), 256, 0, stream>>>(ws, zeroLen);

  // 1) degree + rsqrt
  k_deg<<<(N_EDGES + 255) / 256, 256, 0, stream>>>(dst, deg);
  k_rsqrt<<<(N_NODES + 255) / 256, 256, 0, stream>>>(deg, disq);

  // 2) pre-MLP: h = x @ pre_w + pre_b     (WMMA fp32, K=128)
  k_gemm64<<<(N_NODES + 15) / 16, 128, 0, stream>>>(x, pre_w, pre_b, h, N_NODES, F_IN);

  // 3) conv1: hw = h @ c1_w (no bias)     (WMMA fp32, K=64)
  k_gemm64<<<(N_NODES + 15) / 16, 128, 0, stream>>>(h, c1_w, nullptr, hw, N_NODES, H_DIM);
  k_edge<<<(unsigned)(((long)N_EDGES * 16 + 255) / 256), 256, 0, stream>>>(src, dst, hw, disq, bufA);
  k_selfln<<<(N_NODES + 7) / 8, 256, 0, stream>>>(bufA, hw, disq, c1_b, n1_g, n1_b, bufA);

  // 4) conv2: hw = h1 @ c2_w
  k_gemm64<<<(N_NODES + 15) / 16, 128, 0, stream>>>(bufA, c2_w, nullptr, hw, N_NODES, H_DIM);
  k_edge<<<(unsigned)(((long)N_EDGES * 16 + 255) / 256), 256, 0, stream>>>(src, dst, hw, disq, bufB);
  k_selfln<<<(N_NODES + 7) / 8, 256, 0, stream>>>(bufB, hw, disq, c2_b, n2_g, n2_b, bufB);

  // 5) skip sum + pooling
  k_readout<<<(unsigned)(((long)N_NODES * H_DIM + 255) / 256), 256, 0, stream>>>(bufA, bufB, batch, pooled);

  // 6) head
  k_post<<<(G_GRAPHS * C_OUT + 255) / 256, 256, 0, stream>>>(pooled, post_w, post_b, out);
}